// MattingSolver_16707422781579
// MI455X (gfx1250) — compile-verified
//
#include <hip/hip_runtime.h>

// ---------------------------------------------------------------------------
// CG solve of A x = b, A = implicit fixed-width COO (K=24 off-diag per row,
// contiguous, + diagonal appended). 30 fixed steps, fp32 throughout.
// Memory-bound: ~155 MB/step, L2-resident (192 MB) after first pass.
// WMMA (v_wmma_f32_16x16x4_f32) used for all wave-level dot-product reductions.
// ---------------------------------------------------------------------------

#define CG_N     786432
#define CG_K     24
#define CG_STEPS 30
#define TPB      256
#define NBLK     (CG_N / TPB)   // 3072
#define NWAVES   (TPB / 32)     // 8 waves per block (wave32)

typedef __attribute__((ext_vector_type(2))) float v2f;
typedef __attribute__((ext_vector_type(8))) float v8f;

// Wave32 sum via V_WMMA_F32_16X16X4_F32:
// A (16x4 f32, 2 VGPRs/lane): VGPR0 = lane value s, VGPR1 = 0.
//   => A[m][0] = s_m (m = lane 0..15), A[m][2] = s_{m+16}, rest 0.
// B (4x16) = all ones  => D[m][n] = s_m + s_{m+16} for every n.
// Sum of a lane's 8 D VGPRs t_n covers rows 0..7 (lanes 0-15) / 8..15 (lanes 16-31),
// so total = t(lane 0) + t(lane 16). EXEC must be all ones at the call site.
__device__ __forceinline__ float wave_sum_wmma(float s) {
    v2f a; a[0] = s;    a[1] = 0.0f;
    v2f bm; bm[0] = 1.0f; bm[1] = 1.0f;
    v8f c = {};
    v8f d = __builtin_amdgcn_wmma_f32_16x16x4_f32(
        /*neg_a=*/false, a, /*neg_b=*/false, bm,
        /*c_mod=*/(short)0, c, /*reuse_a=*/false, /*reuse_b=*/false);
    float t = d[0] + d[1] + d[2] + d[3] + d[4] + d[5] + d[6] + d[7];
    return __shfl(t, 0, 32) + __shfl(t, 16, 32);
}

// Block-level sum; result valid in thread 0. All 256 threads must be active.
__device__ __forceinline__ float block_sum(float s, float* sdata) {
    float w = wave_sum_wmma(s);
    int lane = threadIdx.x & 31;
    int wid  = threadIdx.x >> 5;
    if (lane == 0) sdata[wid] = w;
    __syncthreads();
    float total = 0.0f;
    if (threadIdx.x == 0) {
#pragma unroll
        for (int i = 0; i < NWAVES; ++i) total += sdata[i];
    }
    return total;
}

// x = 0, r = p = b, partial[blk] = block sum of b*b
__global__ void cg_init_kernel(const float* __restrict__ b,
                               float* __restrict__ x,
                               float* __restrict__ r,
                               float* __restrict__ p,
                               float* __restrict__ partial) {
    __shared__ float sdata[NWAVES];
    int i = blockIdx.x * TPB + threadIdx.x;
    float bi = b[i];
    x[i] = 0.0f;
    r[i] = bi;
    p[i] = bi;
    float tot = block_sum(bi * bi, sdata);
    if (threadIdx.x == 0) partial[blockIdx.x] = tot;
}

// Ap = A*p (implicit structure), partial[blk] = block sum of p*Ap
__global__ void cg_spmv_dot_kernel(const int*   __restrict__ cols,
                                   const float* __restrict__ vals,
                                   const float* __restrict__ p,
                                   float* __restrict__ Ap,
                                   float* __restrict__ partial) {
    __shared__ float sdata[NWAVES];
    int row = blockIdx.x * TPB + threadIdx.x;
    long base = (long)row * CG_K;                  // byte offset row*96: 16B aligned
    const int4*   c4 = (const int4*)(cols + base);
    const float4* v4 = (const float4*)(vals + base);

    // Uniform-branch prefetch of next block's 96B-per-thread slices
    // (keeps EXEC fully set; emits global_prefetch_b8).
    if (blockIdx.x + 1 < NBLK) {
        __builtin_prefetch(cols + base + (long)TPB * CG_K, 0, 1);
        __builtin_prefetch(vals + base + (long)TPB * CG_K, 0, 1);
    }

    float pr  = p[row];
    float acc = vals[(long)CG_N * CG_K + row] * pr;   // diagonal term
#pragma unroll
    for (int q = 0; q < CG_K / 4; ++q) {
        int4   ci = c4[q];
        float4 vi = v4[q];
        acc += vi.x * p[ci.x];
        acc += vi.y * p[ci.y];
        acc += vi.z * p[ci.z];
        acc += vi.w * p[ci.w];
    }
    Ap[row] = acc;

    float tot = block_sum(pr * acc, sdata);
    if (threadIdx.x == 0) partial[blockIdx.x] = tot;
}

// x += alpha*p ; r -= alpha*Ap ; partial[blk] = block sum of r_new^2
__global__ void cg_update_xr_kernel(float* __restrict__ x,
                                    float* __restrict__ r,
                                    const float* __restrict__ p,
                                    const float* __restrict__ Ap,
                                    const float* __restrict__ scal,
                                    float* __restrict__ partial) {
    __shared__ float sdata[NWAVES];
    int i = blockIdx.x * TPB + threadIdx.x;
    float alpha = scal[2];
    x[i] = x[i] + alpha * p[i];
    float rn = r[i] - alpha * Ap[i];
    r[i] = rn;
    float tot = block_sum(rn * rn, sdata);
    if (threadIdx.x == 0) partial[blockIdx.x] = tot;
}

// p = r + beta*p
__global__ void cg_update_p_kernel(float* __restrict__ p,
                                   const float* __restrict__ r,
                                   const float* __restrict__ scal) {
    int i = blockIdx.x * TPB + threadIdx.x;
    float beta = scal[3];
    p[i] = r[i] + beta * p[i];
}

// Single-block deterministic reduction of the 3072 partials + scalar update.
// mode 0: rs = tot          (after init)
// mode 1: pAp = tot, alpha = rs / pAp
// mode 2: beta = tot / rs, rs = tot
__global__ void cg_reduce_kernel(const float* __restrict__ partial,
                                 float* __restrict__ scal, int mode) {
    __shared__ float sdata[NWAVES];
    float s = 0.0f;
    // NBLK (3072) is a multiple of TPB (256): loop is wave-uniform, EXEC stays full.
    for (int i = threadIdx.x; i < NBLK; i += TPB) s += partial[i];
    float tot = block_sum(s, sdata);
    if (threadIdx.x == 0) {
        if (mode == 0) {
            scal[0] = tot;                    // rs
        } else if (mode == 1) {
            scal[1] = tot;                    // pAp
            scal[2] = scal[0] / tot;          // alpha
        } else {
            scal[3] = tot / scal[0];          // beta
            scal[0] = tot;                    // rs = rs_new
        }
    }
}

extern "C" void kernel_launch(void* const* d_in, const int* in_sizes, int n_in,
                              void* d_out, int out_size, void* d_ws, size_t ws_size,
                              hipStream_t stream) {
    (void)in_sizes; (void)n_in; (void)out_size; (void)ws_size;
    // setup_inputs order: rows(unused), cols, vals, b
    const int*   cols = (const int*)d_in[1];
    const float* vals = (const float*)d_in[2];
    const float* b    = (const float*)d_in[3];
    float* x  = (float*)d_out;

    float* ws      = (float*)d_ws;
    float* Ap      = ws;                     // N floats
    float* r       = ws + (long)CG_N;        // N floats
    float* p       = ws + 2L * CG_N;         // N floats
    float* partial = ws + 3L * CG_N;         // NBLK floats (pad to 4096)
    float* scal    = ws + 3L * CG_N + 4096;  // [rs, pAp, alpha, beta]

    dim3 grid(NBLK), block(TPB);
    cg_init_kernel<<<grid, block, 0, stream>>>(b, x, r, p, partial);
    cg_reduce_kernel<<<1, TPB, 0, stream>>>(partial, scal, 0);

    for (int s = 0; s < CG_STEPS; ++s) {
        cg_spmv_dot_kernel<<<grid, block, 0, stream>>>(cols, vals, p, Ap, partial);
        cg_reduce_kernel<<<1, TPB, 0, stream>>>(partial, scal, 1);
        cg_update_xr_kernel<<<grid, block, 0, stream>>>(x, r, p, Ap, scal, partial);
        cg_reduce_kernel<<<1, TPB, 0, stream>>>(partial, scal, 2);
        cg_update_p_kernel<<<grid, block, 0, stream>>>(p, r, scal);
    }
}